// DynamicOptimizerModule_25417616457970
// MI455X (gfx1250) — compile-verified
//
#include <hip/hip_runtime.h>

// CDNA5 / gfx1250. Workload: fused 8-input -> 4-output per-element linear
// combine over 2048x4096 f32 planes. HBM-bound (384 MiB/call, ~16.5 us floor
// at 23.3 TB/s). Hot loop = b128 non-temporal streaming. The 4x8 coefficient
// matrix (the genome graph collapse) is computed on the matrix pipe with two
// chained V_WMMA_F32_16X16X4_F32 ops (once per wave, amortized to ~0 cost).

typedef float v2f __attribute__((ext_vector_type(2)));
typedef float v4f __attribute__((ext_vector_type(4)));
typedef float v8f __attribute__((ext_vector_type(8)));

__device__ __forceinline__ float rdlane(float v, int lane) {
    return __builtin_bit_cast(
        float, __builtin_amdgcn_readlane(__builtin_bit_cast(int, v), lane));
}

__global__ __launch_bounds__(256)
void genome_combine_kernel(const float* __restrict__ p0, const float* __restrict__ p1,
                           const float* __restrict__ p2, const float* __restrict__ p3,
                           const float* __restrict__ p4, const float* __restrict__ p5,
                           const float* __restrict__ p6, const float* __restrict__ p7,
                           const float* __restrict__ w,
                           float* __restrict__ out, int n4)
{
    // ---------------- coefficient matmul on the WMMA pipe ----------------
    // C(4x8) = O1(4x4)*I1(4x8) + O2(4x4)*I2(4x8), padded into 16x16x4 tiles.
    // Executed with EXEC all-ones (full wave, no divergence yet).
    float ww[24];
#pragma unroll
    for (int i = 0; i < 24; ++i) ww[i] = w[i];

    const int lane = (int)(threadIdx.x & 31u);
    const int m  = lane & 15;   // A: row M / B: col N
    const int hi = lane >> 4;   // K half: A/B VGPR0 holds K={0,2}, VGPR1 K={1,3}

    // A1[o][j]: outer weights for mids 10..13 -> outs 18,19
    v2f a1, a2, b1, b2;
    a1.x = (hi == 0) ? (m == 0 ? ww[16] : 0.f) : (m == 1 ? ww[18] : 0.f); // K=0 / K=2
    a1.y = (hi == 0) ? (m == 0 ? ww[17] : 0.f) : (m == 1 ? ww[19] : 0.f); // K=1 / K=3
    // A2[o][j]: outer weights for mids 14..17 -> outs 20,21
    a2.x = (hi == 0) ? (m == 2 ? ww[20] : 0.f) : (m == 3 ? ww[22] : 0.f);
    a2.y = (hi == 0) ? (m == 2 ? ww[21] : 0.f) : (m == 3 ? ww[23] : 0.f);
    // B1[j][k]: inner weights, mid j over p-index k (mids 10..13)
    b1.x = (hi == 0) ? (m == 0 ? ww[0] : m == 1 ? ww[1] : 0.f)   // J=0: p0*w0+p1*w1
                     : (m == 4 ? ww[4] : m == 5 ? ww[5] : 0.f);  // J=2: p4*w4+p5*w5
    b1.y = (hi == 0) ? (m == 2 ? ww[2] : m == 3 ? ww[3] : 0.f)   // J=1: p2*w2+p3*w3
                     : (m == 6 ? ww[6] : m == 7 ? ww[7] : 0.f);  // J=3: p6*w6+p7*w7
    // B2[j][k]: mids 14..17
    b2.x = (hi == 0) ? (m == 0 ? ww[8]  : m == 7 ? ww[9]  : 0.f)  // J=0: p0*w8+p7*w9
                     : (m == 2 ? ww[12] : m == 5 ? ww[13] : 0.f); // J=2: p2*w12+p5*w13
    b2.y = (hi == 0) ? (m == 1 ? ww[10] : m == 6 ? ww[11] : 0.f)  // J=1: p1*w10+p6*w11
                     : (m == 3 ? ww[14] : m == 4 ? ww[15] : 0.f); // J=3: p3*w14+p4*w15

    v8f d = {};
    d = __builtin_amdgcn_wmma_f32_16x16x4_f32(false, a1, false, b1, (short)0, d, false, false);
    d = __builtin_amdgcn_wmma_f32_16x16x4_f32(false, a2, false, b2, (short)0, d, false, false);

    // D row o lives in VGPR o, N=k in lane k. Broadcast the 16 live coeffs.
    const float c00 = rdlane(d[0], 0), c01 = rdlane(d[0], 1),
                c02 = rdlane(d[0], 2), c03 = rdlane(d[0], 3);
    const float c14 = rdlane(d[1], 4), c15 = rdlane(d[1], 5),
                c16 = rdlane(d[1], 6), c17 = rdlane(d[1], 7);
    const float c20 = rdlane(d[2], 0), c21 = rdlane(d[2], 1),
                c26 = rdlane(d[2], 6), c27 = rdlane(d[2], 7);
    const float c32 = rdlane(d[3], 2), c33 = rdlane(d[3], 3),
                c34 = rdlane(d[3], 4), c35 = rdlane(d[3], 5);

    // ---------------- HBM streaming: fused combine, b128 NT ----------------
    const v4f* __restrict__ P0 = (const v4f*)p0;
    const v4f* __restrict__ P1 = (const v4f*)p1;
    const v4f* __restrict__ P2 = (const v4f*)p2;
    const v4f* __restrict__ P3 = (const v4f*)p3;
    const v4f* __restrict__ P4 = (const v4f*)p4;
    const v4f* __restrict__ P5 = (const v4f*)p5;
    const v4f* __restrict__ P6 = (const v4f*)p6;
    const v4f* __restrict__ P7 = (const v4f*)p7;
    v4f* __restrict__ O = (v4f*)out;

    const size_t N4 = (size_t)n4;
    const size_t stride = (size_t)gridDim.x * blockDim.x;
    for (size_t i = (size_t)blockIdx.x * blockDim.x + threadIdx.x; i < N4; i += stride) {
        v4f x0 = __builtin_nontemporal_load(P0 + i);
        v4f x1 = __builtin_nontemporal_load(P1 + i);
        v4f x2 = __builtin_nontemporal_load(P2 + i);
        v4f x3 = __builtin_nontemporal_load(P3 + i);
        v4f x4 = __builtin_nontemporal_load(P4 + i);
        v4f x5 = __builtin_nontemporal_load(P5 + i);
        v4f x6 = __builtin_nontemporal_load(P6 + i);
        v4f x7 = __builtin_nontemporal_load(P7 + i);

        v4f y0 = c00 * x0 + c01 * x1 + c02 * x2 + c03 * x3;  // out18
        v4f y1 = c14 * x4 + c15 * x5 + c16 * x6 + c17 * x7;  // out19
        v4f y2 = c20 * x0 + c21 * x1 + c26 * x6 + c27 * x7;  // out20
        v4f y3 = c32 * x2 + c33 * x3 + c34 * x4 + c35 * x5;  // out21

        __builtin_nontemporal_store(y0, O + i);
        __builtin_nontemporal_store(y1, O + N4 + i);
        __builtin_nontemporal_store(y2, O + 2 * N4 + i);
        __builtin_nontemporal_store(y3, O + 3 * N4 + i);
    }
}

extern "C" void kernel_launch(void* const* d_in, const int* in_sizes, int n_in,
                              void* d_out, int out_size, void* d_ws, size_t ws_size,
                              hipStream_t stream) {
    // setup_inputs() order: [0]=loss, [1]=prev_loss, [2]=weights(24), [3..10]=p0..p7
    const float* w  = (const float*)d_in[2];
    const float* p0 = (const float*)d_in[3];
    const float* p1 = (const float*)d_in[4];
    const float* p2 = (const float*)d_in[5];
    const float* p3 = (const float*)d_in[6];
    const float* p4 = (const float*)d_in[7];
    const float* p5 = (const float*)d_in[8];
    const float* p6 = (const float*)d_in[9];
    const float* p7 = (const float*)d_in[10];
    float* out = (float*)d_out;

    const int n4 = in_sizes[3] / 4;  // 2048*4096/4 = 2,097,152 float4 per plane

    // 2048 blocks x 256 threads = 16384 wave32 -> 4 float4 iterations/thread;
    // plenty of MLP for 23.3 TB/s, WMMA/readlane setup amortized over 768 B/thread.
    dim3 grid(2048), block(256);
    hipLaunchKernelGGL(genome_combine_kernel, grid, block, 0, stream,
                       p0, p1, p2, p3, p4, p5, p6, p7, w, out, n4);
}